// Decoder_9431748182017
// MI455X (gfx1250) — compile-verified
//
#include <hip/hip_runtime.h>
#include <math.h>

#define Hdim   1024
#define H2dim  512
#define Vdim   32000
#define Wwin   64
#define WINN   129
#define NEGV   (-1e30f)

typedef float v2f __attribute__((ext_vector_type(2)));
typedef float v4f __attribute__((ext_vector_type(4)));
typedef float v8f __attribute__((ext_vector_type(8)));

// D = A(16x4 f32) * B(4x16 f32) + C(16x16 f32), wave32.
__device__ __forceinline__ v8f wmma_k4(v2f a, v2f b, v8f c) {
    return __builtin_amdgcn_wmma_f32_16x16x4_f32(
        /*neg_a=*/false, a, /*neg_b=*/false, b,
        /*c_mod=*/(short)0, c, /*reuse_a=*/false, /*reuse_b=*/false);
}

__device__ __forceinline__ v2f lo2(v4f v) { return __builtin_shufflevector(v, v, 0, 1); }
__device__ __forceinline__ v2f hi2(v4f v) { return __builtin_shufflevector(v, v, 2, 3); }

template <int NT>
__device__ __forceinline__ v4f ld4(const float* p) {
    if constexpr (NT) return __builtin_nontemporal_load((const v4f*)p); // stream, bypass L2 residency
    else              return *(const v4f*)p;                            // keep L2-resident
}

__device__ __forceinline__ float sigm(float x) { return 1.0f / (1.0f + expf(-x)); }

// ---------------- embedding gather ----------------
__global__ void embed_kernel(const float* __restrict__ emb, const int* __restrict__ word,
                             float* __restrict__ x) {
    const int w = word[0];
    for (int k = threadIdx.x; k < Hdim; k += blockDim.x)
        x[k] = emb[(size_t)w * Hdim + k];
}

// ---- WMMA GEMV: y = act(W x + b), 16 rows / wave, b128 loads, batched groups ----
// Each half-wave covers k+4*half..k+4*half+3 per b128; WMMA#1 eats elems {0,1},
// WMMA#2 eats elems {2,3}. A and B share the K-slot mapping, so the permuted
// memory-k -> slot assignment is consistent and the dot product is exact.
template <int ACT, int NT>
__global__ void gemv16_kernel(const float* __restrict__ W, const float* __restrict__ x,
                              const float* __restrict__ bias, float* __restrict__ y,
                              int K) {
    const int lane = threadIdx.x & 31;
    const int wave = (blockIdx.x * blockDim.x + threadIdx.x) >> 5; // one 16-row tile per wave
    const int m    = lane & 15;
    const int koff = (lane >> 4) << 2;          // 0 or 4
    const float* wrow = W + (size_t)(wave * 16 + m) * (size_t)K + koff;
    const float* xp   = x + koff;

    v8f c0 = {}, c1 = {};
    for (int k = 0; k < K; k += 32) {           // group: 8 x b128 loads, then 8 WMMAs
        v4f aw[4], bx[4];
        #pragma unroll
        for (int i = 0; i < 4; ++i) {
            aw[i] = ld4<NT>(wrow + k + 8 * i);
            bx[i] = ld4<0>(xp + k + 8 * i);
        }
        #pragma unroll
        for (int i = 0; i < 4; ++i) {
            c0 = wmma_k4(lo2(aw[i]), lo2(bx[i]), c0);
            c1 = wmma_k4(hi2(aw[i]), hi2(bx[i]), c1);
        }
    }
    v8f c = c0 + c1;
    // C/D layout: lane 0 holds rows 0..7 in vgprs 0..7; lane 16 holds rows 8..15
    if (m == 0) {
        const int rbase = wave * 16 + (lane >> 4) * 8;
        #pragma unroll
        for (int j = 0; j < 8; ++j) {
            float v = c[j] + bias[rbase + j];
            if (ACT == 1) v = tanhf(v);
            y[rbase + j] = v;
        }
    }
}

// ---- dual-matrix WMMA GEMV: y = W1 x1 + W2 x2 + b1 + b2 (LSTM gates), b128 loads ----
__global__ void gemv16_dual_kernel(const float* __restrict__ W1, const float* __restrict__ x1,
                                   const float* __restrict__ W2, const float* __restrict__ x2,
                                   const float* __restrict__ bias1,
                                   const float* __restrict__ bias2,
                                   float* __restrict__ y, int K) {
    const int lane = threadIdx.x & 31;
    const int wave = (blockIdx.x * blockDim.x + threadIdx.x) >> 5;
    const int m    = lane & 15;
    const int koff = (lane >> 4) << 2;
    const float* w1 = W1 + (size_t)(wave * 16 + m) * (size_t)K + koff;
    const float* w2 = W2 + (size_t)(wave * 16 + m) * (size_t)K + koff;
    const float* p1 = x1 + koff;
    const float* p2 = x2 + koff;

    v8f q0 = {}, q1 = {}, q2 = {}, q3 = {};     // four independent chains
    for (int k = 0; k < K; k += 16) {           // group: 8 x b128 loads, then 8 WMMAs
        v4f aA = ld4<0>(w1 + k), aB = ld4<0>(w1 + k + 8);
        v4f cA = ld4<0>(w2 + k), cB = ld4<0>(w2 + k + 8);
        v4f xA = ld4<0>(p1 + k), xB = ld4<0>(p1 + k + 8);
        v4f hA = ld4<0>(p2 + k), hB = ld4<0>(p2 + k + 8);
        q0 = wmma_k4(lo2(aA), lo2(xA), q0);
        q1 = wmma_k4(hi2(aA), hi2(xA), q1);
        q2 = wmma_k4(lo2(cA), lo2(hA), q2);
        q3 = wmma_k4(hi2(cA), hi2(hA), q3);
        q0 = wmma_k4(lo2(aB), lo2(xB), q0);
        q1 = wmma_k4(hi2(aB), hi2(xB), q1);
        q2 = wmma_k4(lo2(cB), lo2(hB), q2);
        q3 = wmma_k4(hi2(cB), hi2(hB), q3);
    }
    v8f c = (q0 + q1) + (q2 + q3);
    if (m == 0) {
        const int rbase = wave * 16 + (lane >> 4) * 8;
        #pragma unroll
        for (int j = 0; j < 8; ++j)
            y[rbase + j] = c[j] + bias1[rbase + j] + bias2[rbase + j];
    }
}

// ---------------- LSTM cell elementwise update ----------------
__global__ void lstm_cell_kernel(const float* __restrict__ gates, const float* __restrict__ cprev,
                                 float* __restrict__ hout, float* __restrict__ cout) {
    const int h = blockIdx.x * blockDim.x + threadIdx.x;   // 0..1023
    const float gi = gates[h];
    const float gf = gates[Hdim + h];
    const float gg = gates[2 * Hdim + h];
    const float go = gates[3 * Hdim + h];
    const float cn = sigm(gf) * cprev[h] + sigm(gi) * tanhf(gg);
    const float hn = sigm(go) * tanhf(cn);
    cout[h] = cn;
    hout[h] = hn;
}

// ---------------- p = Sf * sigmoid(att1 . w2 + b2) (single wave) ----------------
__global__ void att2_p_kernel(const float* __restrict__ att1, const float* __restrict__ w2,
                              const float* __restrict__ b2,   const int* __restrict__ slen,
                              float* __restrict__ pout) {
    const int lane = threadIdx.x;
    float s = 0.f;
    for (int k = lane; k < H2dim; k += 32) s += att1[k] * w2[k];
    #pragma unroll
    for (int off = 16; off; off >>= 1) s += __shfl_xor(s, off, 32);
    if (lane == 0) pout[0] = (float)slen[0] * sigm(s + b2[0]);
}

// ---------------- windowed scores + softmax*gauss + context (one block) ----------------
__global__ void attn_kernel(const float* __restrict__ enc, const float* __restrict__ ht,
                            const float* __restrict__ pws, const int* __restrict__ slen,
                            float* __restrict__ a_out, float* __restrict__ cat) {
    __shared__ float sa[WINN];
    __shared__ float red[256];
    const int t    = threadIdx.x;
    const int lane = t & 31;
    const int wv   = t >> 5;                       // 8 waves
    const int S    = slen[0];
    const float p  = pws[0];
    const int wsI  = (int)rintf(fmaxf(p - (float)Wwin, 0.f));
    const int weI  = (int)rintf(fminf(p + (float)Wwin, (float)(S - 1)));

    // scores[j] = valid ? h_s[j] . h_t : NEG
    for (int j = wv; j < WINN; j += 8) {
        const int idx = wsI + j;
        const int ci  = idx < S - 1 ? idx : S - 1;
        const float* row = enc + (size_t)ci * Hdim;
        float s = 0.f;
        for (int k = lane; k < Hdim; k += 32) s += row[k] * ht[k];
        #pragma unroll
        for (int off = 16; off; off >>= 1) s += __shfl_xor(s, off, 32);
        if (lane == 0) sa[j] = (idx <= weI) ? s : NEGV;
    }
    __syncthreads();
    // max
    red[t] = (t < WINN) ? sa[t] : NEGV;
    __syncthreads();
    for (int off = 128; off; off >>= 1) { if (t < off) red[t] = fmaxf(red[t], red[t + off]); __syncthreads(); }
    const float mx = red[0];
    __syncthreads();
    // sum of exp
    const float e = (t < WINN) ? expf(sa[t] - mx) : 0.f;
    red[t] = e;
    __syncthreads();
    for (int off = 128; off; off >>= 1) { if (t < off) red[t] += red[t + off]; __syncthreads(); }
    const float sum = red[0];
    __syncthreads();
    // a = valid ? softmax * exp((idx-p)/2048) : 0   (matches reference formula verbatim)
    if (t < WINN) {
        const int idx = wsI + t;
        float a = 0.f;
        if (idx <= weI) a = (e / sum) * expf(((float)idx - p) * (1.0f / 2048.0f));
        sa[t] = a;
        a_out[t] = a;
    }
    __syncthreads();
    // ctx = a @ h_s ; cat = [ctx, h_t]
    for (int k = t; k < Hdim; k += 256) {
        float acc = 0.f;
        for (int j = 0; j < WINN; ++j) {
            const int idx = wsI + j;
            const int ci  = idx < S - 1 ? idx : S - 1;
            acc += sa[j] * enc[(size_t)ci * Hdim + k];
        }
        cat[k]        = acc;
        cat[Hdim + k] = ht[k];
    }
}

// ---------------- log-softmax over V (one block) ----------------
__global__ void logsoftmax_kernel(const float* __restrict__ logits, float* __restrict__ y) {
    __shared__ float red[1024];
    const int t = threadIdx.x;
    float mx = NEGV;
    for (int i = t; i < Vdim; i += 1024) mx = fmaxf(mx, logits[i]);
    red[t] = mx;
    __syncthreads();
    for (int off = 512; off; off >>= 1) { if (t < off) red[t] = fmaxf(red[t], red[t + off]); __syncthreads(); }
    mx = red[0];
    __syncthreads();
    float s = 0.f;
    for (int i = t; i < Vdim; i += 1024) s += expf(logits[i] - mx);
    red[t] = s;
    __syncthreads();
    for (int off = 512; off; off >>= 1) { if (t < off) red[t] += red[t + off]; __syncthreads(); }
    const float lse = mx + logf(red[0]);
    for (int i = t; i < Vdim; i += 1024) y[i] = logits[i] - lse;
}

extern "C" void kernel_launch(void* const* d_in, const int* in_sizes, int n_in,
                              void* d_out, int out_size, void* d_ws, size_t ws_size,
                              hipStream_t stream) {
    const int*   slen = (const int*)d_in[0];
    const float* enc  = (const float*)d_in[1];
    const int*   word = (const int*)d_in[2];
    const float* h0   = (const float*)d_in[3];
    const float* c0   = (const float*)d_in[4];
    const float* emb  = (const float*)d_in[5];
    const float* w_ih = (const float*)d_in[6];
    const float* w_hh = (const float*)d_in[7];
    const float* b_ih = (const float*)d_in[8];
    const float* b_hh = (const float*)d_in[9];
    const float* a1w  = (const float*)d_in[10];
    const float* a1b  = (const float*)d_in[11];
    const float* a2w  = (const float*)d_in[12];
    const float* a2b  = (const float*)d_in[13];
    const float* f1w  = (const float*)d_in[14];
    const float* f1b  = (const float*)d_in[15];
    const float* f2w  = (const float*)d_in[16];
    const float* f2b  = (const float*)d_in[17];

    float* out  = (float*)d_out;
    float* y    = out;              // 32000
    float* outv = out + 32000;      // 1024
    float* h_n  = out + 33024;      // 2048 (2 layers)
    float* c_n  = out + 35072;      // 2048
    float* a_o  = out + 37120;      // 129

    float* ws     = (float*)d_ws;
    float* xemb   = ws;             // 1024
    float* att1   = ws + 1024;      // 512
    float* pws    = ws + 1536;      // 1
    float* cat    = ws + 2048;      // 2048 [ctx | h_t]
    float* logits = ws + 4096;      // 32000
    float* gates  = ws + 36096;     // 4096

    embed_kernel<<<1, 256, 0, stream>>>(emb, word, xemb);

    // LSTM layer 0: gates = W_ih x + W_hh h0 + biases  (4096 rows -> 256 waves)
    gemv16_dual_kernel<<<32, 256, 0, stream>>>(w_ih, xemb, w_hh, h0,
                                               b_ih, b_hh, gates, Hdim);
    lstm_cell_kernel<<<4, 256, 0, stream>>>(gates, c0, h_n, c_n);

    // LSTM layer 1
    gemv16_dual_kernel<<<32, 256, 0, stream>>>(w_ih + (size_t)4 * Hdim * Hdim, h_n,
                                               w_hh + (size_t)4 * Hdim * Hdim, h0 + Hdim,
                                               b_ih + 4 * Hdim, b_hh + 4 * Hdim, gates, Hdim);
    lstm_cell_kernel<<<4, 256, 0, stream>>>(gates, c0 + Hdim, h_n + Hdim, c_n + Hdim);

    const float* ht = h_n + Hdim;

    // attention fc1: 512 rows -> 32 tiles -> 4 blocks x 8 waves (L2-resident weights)
    gemv16_kernel<1, 0><<<4, 256, 0, stream>>>(a1w, ht, a1b, att1, Hdim);
    att2_p_kernel<<<1, 32, 0, stream>>>(att1, a2w, a2b, slen, pws);
    attn_kernel<<<1, 256, 0, stream>>>(enc, ht, pws, slen, a_o, cat);

    // fc1: 1024 rows, K=2048 -> 64 tiles (L2-resident weights)
    gemv16_kernel<1, 0><<<8, 256, 0, stream>>>(f1w, cat, f1b, outv, 2 * Hdim);
    // fc2: 32000 rows, K=1024 -> 2000 tiles; 131MB weights can't fit 192MB L2 alongside
    // the resident set -> stream them non-temporally to protect the L2 working set.
    gemv16_kernel<0, 1><<<250, 256, 0, stream>>>(f2w, outv, f2b, logits, Hdim);

    logsoftmax_kernel<<<1, 1024, 0, stream>>>(logits, y);
}